// MultiheadRelationBlock_27711128993860
// MI455X (gfx1250) — compile-verified
//
#include <hip/hip_runtime.h>
#include <hip/hip_bf16.h>

typedef __attribute__((ext_vector_type(2))) float v2f;
typedef __attribute__((ext_vector_type(8))) float v8f;

#define HEADS  8
#define NB     16
#define CH     512
#define HT     64
#define WD     64
#define HWSZ   4096
#define PLANES 128

#if __has_builtin(__builtin_amdgcn_global_load_async_to_lds_b128) && \
    __has_builtin(__builtin_amdgcn_s_wait_asynccnt)
#define USE_ASYNC_LDS 1
#else
#define USE_ASYNC_LDS 0
#endif

// async copy builtin wants typed int4 (vector_size 16) pointers:
//   arg0: AS1 (global, prints as __device__), arg1: AS3 (LDS, prints as __shared__)
typedef int async_v4i __attribute__((vector_size(16)));
#define ASYNC_GPTR(p)                                              \
    ((__attribute__((address_space(1))) async_v4i*)                \
     (__attribute__((address_space(1))) void*)(void*)(p))
#define ASYNC_LPTR(p)                                              \
    ((__attribute__((address_space(3))) async_v4i*)                \
     (__attribute__((address_space(3))) void*)(void*)(p))

// ---------------------------------------------------------------------------
// Kernel 1: logits[b,n,hw] = sum_c x[b,c,hw] * cmw[n,c] + cmb[n]
// WMMA f32 16x16x4: M = heads (8 used of 16), N = 16 hw columns, K over C.
// ---------------------------------------------------------------------------
__global__ void __launch_bounds__(128)
k_logits_wmma(const float* __restrict__ x, const float* __restrict__ cmw,
              const float* __restrict__ cmb, float* __restrict__ mask)
{
    const int wave = threadIdx.x >> 5;
    const int lane = threadIdx.x & 31;
    const int b    = blockIdx.x >> 6;                 // 64 hw-blocks per batch
    const int hw0  = ((blockIdx.x & 63) << 6) + (wave << 4);
    const int col  = lane & 15;                       // M for A-frag, N for B-frag
    const int kgrp = lane >> 4;                       // 0 or 1: K offset group
    const long xbase = (long)b * CH * HWSZ;

    v8f c = {0.f, 0.f, 0.f, 0.f, 0.f, 0.f, 0.f, 0.f};

    for (int k = 0; k < CH; k += 4) {
        const int kk = k + kgrp * 2;
        // A fragment: conv_mask_w[m, kk], rows >= 8 are zero padding
        float a0 = cmw[(col & 7) * CH + kk];
        float a1 = cmw[(col & 7) * CH + kk + 1];
        v2f af; af.x = (col < HEADS) ? a0 : 0.f;
                af.y = (col < HEADS) ? a1 : 0.f;
        // B fragment: x[b, kk, hw0+col] (coalesced across lanes 0-15)
        v2f bf;
        bf.x = x[xbase + (long)kk       * HWSZ + hw0 + col];
        bf.y = x[xbase + (long)(kk + 1) * HWSZ + hw0 + col];
        if (k + 8 < CH)
            __builtin_prefetch(&x[xbase + (long)(k + 8) * HWSZ + hw0 + col], 0, 1);
        c = __builtin_amdgcn_wmma_f32_16x16x4_f32(false, af, false, bf,
                                                  (short)0, c, false, false);
    }

    // C/D layout: VGPR r, lanes 0-15 -> M=r (heads 0..7), lanes 16-31 -> M=r+8 (pad)
    if (lane < 16) {
        const int n = lane;
#pragma unroll
        for (int r = 0; r < HEADS; ++r)
            mask[(long)(b * HEADS + r) * HWSZ + hw0 + n] = c[r] + cmb[r];
    }
}

// ---------------------------------------------------------------------------
// Kernel 2: per-(b,head) softmax over HW (in place) + centroid cx, cy
// ---------------------------------------------------------------------------
__global__ void __launch_bounds__(256)
k_softmax_centroid(float* __restrict__ mask, float* __restrict__ cx,
                   float* __restrict__ cy)
{
    __shared__ float red[256];
    const int row = blockIdx.x;                       // b*8 + n
    const int tid = threadIdx.x;
    float* r = mask + (long)row * HWSZ;

    float v[16];
    float mx = -1e30f;
#pragma unroll
    for (int i = 0; i < 16; ++i) { v[i] = r[i * 256 + tid]; mx = fmaxf(mx, v[i]); }

    red[tid] = mx; __syncthreads();
    for (int s = 128; s > 0; s >>= 1) {
        if (tid < s) red[tid] = fmaxf(red[tid], red[tid + s]);
        __syncthreads();
    }
    mx = red[0]; __syncthreads();

    float sum = 0.f, pc = 0.f, pr = 0.f;
#pragma unroll
    for (int i = 0; i < 16; ++i) {
        float e = expf(v[i] - mx);
        v[i] = e; sum += e;
        int hw = i * 256 + tid;
        pc += e * (float)(hw & 63);                   // column index j
        pr += e * (float)(hw >> 6);                   // row index i
    }
    red[tid] = sum; __syncthreads();
    for (int s = 128; s > 0; s >>= 1) {
        if (tid < s) red[tid] += red[tid + s];
        __syncthreads();
    }
    sum = red[0]; __syncthreads();

    red[tid] = pc; __syncthreads();
    for (int s = 128; s > 0; s >>= 1) {
        if (tid < s) red[tid] += red[tid + s];
        __syncthreads();
    }
    pc = red[0]; __syncthreads();

    red[tid] = pr; __syncthreads();
    for (int s = 128; s > 0; s >>= 1) {
        if (tid < s) red[tid] += red[tid + s];
        __syncthreads();
    }
    pr = red[0];

    float inv = 1.f / sum;
#pragma unroll
    for (int i = 0; i < 16; ++i) r[i * 256 + tid] = v[i] * inv;

    if (tid == 0) {
        cx[row] = pc * inv / (float)HT;
        cy[row] = pr * inv / (float)WD;
    }
}

// ---------------------------------------------------------------------------
// Kernel 3: context[b,h,c] = sum_hw x[b,c,hw] * mask[b,h,hw]
// WMMA f32 16x16x4: M = 16 channels, N = heads (8 used of 16), K over HW.
// x tiles staged into LDS via GLOBAL_LOAD_ASYNC_TO_LDS_B128, double-buffered
// and synced with s_wait_asynccnt (per-wave counter; each of the 4 waves owns
// its own tile buffers, so no block barrier is needed).
// ---------------------------------------------------------------------------
__global__ void __launch_bounds__(128)
k_context_wmma(const float* __restrict__ x, const float* __restrict__ mask,
               float* __restrict__ ctx)
{
    __shared__ float tile[4][2][16][68];              // [wave][buf][c][hw(+pad)]
    const int wave = threadIdx.x >> 5;
    const int lane = threadIdx.x & 31;
    const int b    = blockIdx.x >> 3;                 // 8 blocks per batch
    const int c0   = (((blockIdx.x & 7) << 2) + wave) << 4;   // 32 c-tiles/batch
    const int col  = lane & 15;
    const int kgrp = lane >> 4;
    const long xbase = (long)b * CH * HWSZ;
    const long mbase = (long)b * HEADS * HWSZ;

    v8f acc = {0.f, 0.f, 0.f, 0.f, 0.f, 0.f, 0.f, 0.f};

#if USE_ASYNC_LDS
    // ---- async double-buffered path ----
    // fill chunk 0 into buffer 0
    {
        const int hwc = 0;
#pragma unroll
        for (int i = 0; i < 8; ++i) {
            int idx = i * 32 + lane;
            int row = idx >> 4, c4 = idx & 15;
            __builtin_amdgcn_global_load_async_to_lds_b128(
                ASYNC_GPTR(&x[xbase + (long)(c0 + row) * HWSZ + hwc + c4 * 4]),
                ASYNC_LPTR(&tile[wave][0][row][c4 * 4]), 0, 0);
        }
    }
    for (int cidx = 0; cidx < HWSZ / 64; ++cidx) {
        const int cur = cidx & 1;
        const int hwc = cidx << 6;
        if (cidx + 1 < HWSZ / 64) {
            const int hwn = hwc + 64;
#pragma unroll
            for (int i = 0; i < 8; ++i) {
                int idx = i * 32 + lane;
                int row = idx >> 4, c4 = idx & 15;
                __builtin_amdgcn_global_load_async_to_lds_b128(
                    ASYNC_GPTR(&x[xbase + (long)(c0 + row) * HWSZ + hwn + c4 * 4]),
                    ASYNC_LPTR(&tile[wave][cur ^ 1][row][c4 * 4]), 0, 0);
            }
            __builtin_amdgcn_s_wait_asynccnt(8);      // chunk cidx complete
        } else {
            __builtin_amdgcn_s_wait_asynccnt(0);
        }
        __builtin_amdgcn_wave_barrier();

        for (int k = 0; k < 64; k += 4) {
            const int kk = k + kgrp * 2;
            v2f af; af.x = tile[wave][cur][col][kk];
                    af.y = tile[wave][cur][col][kk + 1];
            float m0 = mask[mbase + (long)(col & 7) * HWSZ + hwc + kk];
            float m1 = mask[mbase + (long)(col & 7) * HWSZ + hwc + kk + 1];
            v2f bf; bf.x = (col < HEADS) ? m0 : 0.f;
                    bf.y = (col < HEADS) ? m1 : 0.f;
            acc = __builtin_amdgcn_wmma_f32_16x16x4_f32(false, af, false, bf,
                                                        (short)0, acc, false, false);
        }
        __builtin_amdgcn_wave_barrier();
    }
#else
    // ---- fallback: register-staged LDS fill (per-wave, in-order DS) ----
    for (int hwc = 0; hwc < HWSZ; hwc += 64) {
#pragma unroll
        for (int i = 0; i < 8; ++i) {
            int idx = i * 32 + lane;
            int row = idx >> 4, c4 = idx & 15;
            float4 val = *reinterpret_cast<const float4*>(
                &x[xbase + (long)(c0 + row) * HWSZ + hwc + c4 * 4]);
            tile[wave][0][row][c4 * 4 + 0] = val.x;
            tile[wave][0][row][c4 * 4 + 1] = val.y;
            tile[wave][0][row][c4 * 4 + 2] = val.z;
            tile[wave][0][row][c4 * 4 + 3] = val.w;
        }
        __builtin_amdgcn_wave_barrier();

        for (int k = 0; k < 64; k += 4) {
            const int kk = k + kgrp * 2;
            v2f af; af.x = tile[wave][0][col][kk];
                    af.y = tile[wave][0][col][kk + 1];
            float m0 = mask[mbase + (long)(col & 7) * HWSZ + hwc + kk];
            float m1 = mask[mbase + (long)(col & 7) * HWSZ + hwc + kk + 1];
            v2f bf; bf.x = (col < HEADS) ? m0 : 0.f;
                    bf.y = (col < HEADS) ? m1 : 0.f;
            acc = __builtin_amdgcn_wmma_f32_16x16x4_f32(false, af, false, bf,
                                                        (short)0, acc, false, false);
        }
        __builtin_amdgcn_wave_barrier();
    }
#endif

    // D layout: VGPR r, lanes 0-15 -> M=r, lanes 16-31 -> M=r+8; N = lane&15 (head)
    const int h = lane & 15;
    if (h < HEADS) {
        const int mofs = (lane >> 4) * 8;
#pragma unroll
        for (int r = 0; r < 8; ++r)
            ctx[(long)(b * HEADS + h) * CH + c0 + r + mofs] = acc[r];
    }
}

// ---------------------------------------------------------------------------
// Kernel 4: per-batch head attention + channel_add MLP -> add_term[b, 512]
// ---------------------------------------------------------------------------
__global__ void __launch_bounds__(256)
k_head_attn(const float* __restrict__ ctx, const float* __restrict__ cx,
            const float* __restrict__ cy,
            const float* __restrict__ g_w, const float* __restrict__ g_b,
            const float* __restrict__ k_w, const float* __restrict__ k_b,
            const float* __restrict__ q_w, const float* __restrict__ q_b,
            const float* __restrict__ v_w, const float* __restrict__ v_b,
            const float* __restrict__ w1,  const float* __restrict__ b1,
            const float* __restrict__ lng, const float* __restrict__ lnb,
            const float* __restrict__ w2,  const float* __restrict__ b2,
            float* __restrict__ addt)
{
    __shared__ float wkqv[3 * 512];                   // wk | wq | wv, each [h][d]
    __shared__ float lt[64];
    __shared__ float wmn[64];
    __shared__ float oc[512];
    __shared__ float ly[128];
    __shared__ float stats[2];

    const int b   = blockIdx.x;
    const int tid = threadIdx.x;
    const float* cb = ctx + (long)b * HEADS * CH;

    // w_k / w_q / w_v : [8, 64] each, dot over C=512
    for (int o = tid; o < 3 * 512; o += 256) {
        int mat = o >> 9, rem = o & 511, h = rem >> 6, d = rem & 63;
        const float* wm = (mat == 0) ? k_w : (mat == 1) ? q_w : v_w;
        const float* bs = (mat == 0) ? k_b : (mat == 1) ? q_b : v_b;
        float s = bs[d];
        const float* cr = cb + h * CH;
        const float* wr = wm + d * CH;
        for (int c = 0; c < CH; ++c) s += cr[c] * wr[c];
        wkqv[o] = s;
    }
    __syncthreads();

    // w_g (geometric) and scaled_dot -> logits lt[m*8+n]
    if (tid < 64) {
        int m = tid >> 3, n = tid & 7;
        float dx = cx[b * 8 + m] - cx[b * 8 + n];
        float dy = cy[b * 8 + m] - cy[b * 8 + n];
        float pos[4] = { fmaxf(dx, 0.f), fmaxf(-dx, 0.f),
                         fmaxf(dy, 0.f), fmaxf(-dy, 0.f) };
        float s = g_b[0];
        for (int p = 0; p < 4; ++p)
            for (int f = 0; f < 8; ++f) {
                float freq = expf(-6.90775527898f * (float)f * 0.125f); // 1000^(-f/8)
                float ang  = 100.f * pos[p] * freq;
                s += sinf(ang) * g_w[p * 8 + f] + cosf(ang) * g_w[32 + p * 8 + f];
            }
        float wg = fmaxf(s, 0.f);
        float sd = 0.f;
        for (int d = 0; d < 64; ++d) sd += wkqv[m * 64 + d] * wkqv[512 + n * 64 + d];
        sd *= 0.125f;                                 // 1/sqrt(64)
        lt[tid] = logf(fmaxf(wg, 1e-6f)) + sd;
    }
    __syncthreads();

    // softmax over m (axis=1) for each n
    if (tid < 8) {
        int n = tid;
        float mx = -1e30f;
        for (int m = 0; m < 8; ++m) mx = fmaxf(mx, lt[m * 8 + n]);
        float s = 0.f;
        for (int m = 0; m < 8; ++m) s += expf(lt[m * 8 + n] - mx);
        float inv = 1.f / s;
        for (int m = 0; m < 8; ++m) wmn[m * 8 + n] = expf(lt[m * 8 + n] - mx) * inv;
    }
    __syncthreads();

    // out_ctx[n, d] = sum_m wmn[m,n] * wv[m,d] -> flat n*64+d
    for (int o = tid; o < 512; o += 256) {
        int n = o >> 6, d = o & 63;
        float s = 0.f;
        for (int m = 0; m < 8; ++m) s += wmn[m * 8 + n] * wkqv[1024 + m * 64 + d];
        oc[o] = s;
    }
    __syncthreads();

    // y = oc @ w1^T + b1  [128]
    if (tid < PLANES) {
        float s = b1[tid];
        const float* wr = w1 + tid * 512;
        for (int j = 0; j < 512; ++j) s += oc[j] * wr[j];
        ly[tid] = s;
    }
    __syncthreads();

    if (tid == 0) {
        float mu = 0.f;
        for (int p = 0; p < PLANES; ++p) mu += ly[p];
        mu *= (1.f / PLANES);
        float var = 0.f;
        for (int p = 0; p < PLANES; ++p) { float d = ly[p] - mu; var += d * d; }
        var *= (1.f / PLANES);
        stats[0] = mu;
        stats[1] = rsqrtf(var + 1e-5f);
    }
    __syncthreads();

    if (tid < PLANES)
        ly[tid] = fmaxf((ly[tid] - stats[0]) * stats[1] * lng[tid] + lnb[tid], 0.f);
    __syncthreads();

    // add_term[c] = y @ w2^T + b2  [512]
    for (int c = tid; c < 512; c += 256) {
        float s = b2[c];
        const float* wr = w2 + c * PLANES;
        for (int p = 0; p < PLANES; ++p) s += ly[p] * wr[p];
        addt[b * 512 + c] = s;
    }
}

// ---------------------------------------------------------------------------
// Kernel 5: out[b,c,hw] = x[b,c,hw] + add_term[b,c]   (float4 streaming)
// ---------------------------------------------------------------------------
__global__ void __launch_bounds__(256)
k_addout(const float* __restrict__ x, const float* __restrict__ addt,
         float* __restrict__ out)
{
    long e = (long)blockIdx.x * 256 + threadIdx.x;    // float4 index
    float a = addt[e >> 10];                          // 1024 float4 per (b,c) plane
    float4 xv = reinterpret_cast<const float4*>(x)[e];
    float4 o = { xv.x + a, xv.y + a, xv.z + a, xv.w + a };
    reinterpret_cast<float4*>(out)[e] = o;
}

// ---------------------------------------------------------------------------
extern "C" void kernel_launch(void* const* d_in, const int* in_sizes, int n_in,
                              void* d_out, int out_size, void* d_ws, size_t ws_size,
                              hipStream_t stream)
{
    const float* x   = (const float*)d_in[0];
    const float* cmw = (const float*)d_in[1];
    const float* cmb = (const float*)d_in[2];
    const float* g_w = (const float*)d_in[3];
    const float* g_b = (const float*)d_in[4];
    const float* k_w = (const float*)d_in[5];
    const float* k_b = (const float*)d_in[6];
    const float* q_w = (const float*)d_in[7];
    const float* q_b = (const float*)d_in[8];
    const float* v_w = (const float*)d_in[9];
    const float* v_b = (const float*)d_in[10];
    const float* w1  = (const float*)d_in[11];
    const float* b1  = (const float*)d_in[12];
    const float* lng = (const float*)d_in[13];
    const float* lnb = (const float*)d_in[14];
    const float* w2  = (const float*)d_in[15];
    const float* b2  = (const float*)d_in[16];
    float* out = (float*)d_out;

    // workspace layout (floats)
    float* ws   = (float*)d_ws;
    float* mask = ws;                                 // 16*8*4096 = 524288
    float* ctx  = ws + 524288;                        // 16*8*512  = 65536
    float* cxb  = ws + 524288 + 65536;                // 128
    float* cyb  = cxb + 128;                          // 128
    float* addt = cyb + 128;                          // 16*512 = 8192

    k_logits_wmma<<<dim3(NB * 64), dim3(128), 0, stream>>>(x, cmw, cmb, mask);
    k_softmax_centroid<<<dim3(NB * HEADS), dim3(256), 0, stream>>>(mask, cxb, cyb);
    k_context_wmma<<<dim3(NB * 8), dim3(128), 0, stream>>>(x, mask, ctx);
    k_head_attn<<<dim3(NB), dim3(256), 0, stream>>>(ctx, cxb, cyb, g_w, g_b,
                                                    k_w, k_b, q_w, q_b, v_w, v_b,
                                                    w1, b1, lng, lnb, w2, b2, addt);
    k_addout<<<dim3(NB * CH * HWSZ / 4 / 256), dim3(256), 0, stream>>>(x, addt, out);
}